// SanctionImpactGNN_22900765623077
// MI455X (gfx1250) — compile-verified
//
#include <hip/hip_runtime.h>
#include <math.h>

// ---------------------------------------------------------------------------
// SanctionImpactGNN on MI455X (gfx1250, wave32)
//   T=8, N=20000, E=320000, F_IN=128, H=64, INDIA=0
// ---------------------------------------------------------------------------

#define T_STEPS 8
#define NN 20000
#define EE 320000
#define FIN 128
#define HH 64

typedef __attribute__((ext_vector_type(2))) float v2f;
typedef __attribute__((ext_vector_type(8))) float v8f;

// ---- generic fill ---------------------------------------------------------
__global__ void k_fill(float* __restrict__ p, float v, int n) {
  int i = blockIdx.x * blockDim.x + threadIdx.x;
  if (i < n) p[i] = v;
}

// ---- degree: deg[dst] += ew  (deg pre-initialized to 1.0 for self loop) ---
__global__ void k_deg(const int* __restrict__ dst, const float* __restrict__ ew,
                      float* __restrict__ deg, int E) {
  int e = blockIdx.x * blockDim.x + threadIdx.x;
  if (e < E) atomicAdd(&deg[dst[e]], ew[e]);
}

// ---- dis = rsqrt(max(deg, 1e-12)) in place --------------------------------
__global__ void k_rsqrt(float* __restrict__ d, int n) {
  int i = blockIdx.x * blockDim.x + threadIdx.x;
  if (i < n) d[i] = rsqrtf(fmaxf(d[i], 1e-12f));
}

// ---- norm[e] = dis[src]*ew*dis[dst] ---------------------------------------
__global__ void k_norm(const int* __restrict__ src, const int* __restrict__ dst,
                       const float* __restrict__ ew, const float* __restrict__ dis,
                       float* __restrict__ nrm, int E) {
  int e = blockIdx.x * blockDim.x + threadIdx.x;
  if (e < E) nrm[e] = dis[src[e]] * ew[e] * dis[dst[e]];
}

// ---- WMMA f32 GEMM: C[M,64] = A[M,K] @ W[K,64] ----------------------------
// One wave computes a 16-row x 64-col tile with 4 accumulators, stepping K by 4
// via V_WMMA_F32_16X16X4_F32 (full f32 precision).
// A 16x4 frag:  lane l (mi=l&15, h=l>>4): elem v -> A[m0+mi][k0 + 2h + v]
// B 4x16 frag:  lane l: elem v -> W[k0 + 2h + v][n0 + mi]
// C 16x16 frag: elem v -> C[m0 + v + 8h][n0 + mi]
__global__ void __launch_bounds__(128) k_gemm_wmma(
    const float* __restrict__ A, const float* __restrict__ W,
    float* __restrict__ C, int M, int K) {
  int wid  = (blockIdx.x * blockDim.x + threadIdx.x) >> 5;
  int lane = threadIdx.x & 31;
  int m0 = wid * 16;
  if (m0 >= M) return;               // wave-uniform: EXEC stays all-ones
  int h  = lane >> 4;                // lane half
  int mi = lane & 15;

  const float* Arow = A + (size_t)(m0 + mi) * K;
  v8f acc0 = {}, acc1 = {}, acc2 = {}, acc3 = {};

  for (int k0 = 0; k0 < K; k0 += 4) {
    int ka = k0 + 2 * h;
    v2f a;
    a.x = Arow[ka];
    a.y = Arow[ka + 1];
    const float* W0 = W + (size_t)ka * HH + mi;   // row ka
    const float* W1r = W0 + HH;                   // row ka+1
    v2f b0; b0.x = W0[0];  b0.y = W1r[0];
    v2f b1; b1.x = W0[16]; b1.y = W1r[16];
    v2f b2; b2.x = W0[32]; b2.y = W1r[32];
    v2f b3; b3.x = W0[48]; b3.y = W1r[48];
    acc0 = __builtin_amdgcn_wmma_f32_16x16x4_f32(false, a, false, b0, (short)0, acc0, false, false);
    acc1 = __builtin_amdgcn_wmma_f32_16x16x4_f32(false, a, false, b1, (short)0, acc1, false, false);
    acc2 = __builtin_amdgcn_wmma_f32_16x16x4_f32(false, a, false, b2, (short)0, acc2, false, false);
    acc3 = __builtin_amdgcn_wmma_f32_16x16x4_f32(false, a, false, b3, (short)0, acc3, false, false);
  }

#pragma unroll
  for (int v = 0; v < 8; ++v) {
    float* cr = C + (size_t)(m0 + v + 8 * h) * HH + mi;
    cr[0]  = acc0[v];
    cr[16] = acc1[v];
    cr[32] = acc2[v];
    cr[48] = acc3[v];
  }
}

// ---- scatter: agg[dst][c] += h[src][c] * norm[e], thread = e*64+c ---------
__global__ void k_scatter(const float* __restrict__ h, const int* __restrict__ src,
                          const int* __restrict__ dst, const float* __restrict__ nrm,
                          float* __restrict__ agg, int E) {
  long long i = (long long)blockIdx.x * blockDim.x + threadIdx.x;
  long long tot = (long long)E * HH;
  if (i >= tot) return;
  int e = (int)(i >> 6);
  int c = (int)(i & 63);
  float v = h[(long long)src[e] * HH + c] * nrm[e];
  atomicAdd(&agg[(long long)dst[e] * HH + c], v);
}

// ---- layer-1 finalize (in place into agg):
//      h1 = relu(agg + dis[n]^2 * h1g + b1)  (self-loop norm = dis^2) -------
__global__ void k_fin1(float* __restrict__ agg, const float* __restrict__ h1g,
                       const float* __restrict__ dis, const float* __restrict__ b1,
                       int N) {
  int i = blockIdx.x * blockDim.x + threadIdx.x;
  if (i >= N * HH) return;
  int n = i >> 6, c = i & 63;
  float d = dis[n];
  float v = agg[i] + d * d * h1g[i] + b1[c];
  agg[i] = v > 0.f ? v : 0.f;
}

// ---- layer-2 pre-GEMM reduction restricted to node 0 (linearity):
//      vacc[c] += h1[src][c]*norm[e]  for edges with dst==0 -----------------
__global__ void k_red0(const float* __restrict__ h1, const int* __restrict__ src,
                       const int* __restrict__ dst, const float* __restrict__ nrm,
                       float* __restrict__ vacc, int E) {
  int e = blockIdx.x * blockDim.x + threadIdx.x;
  if (e >= E) return;
  if (dst[e] == 0) {
    float nm = nrm[e];
    const float* hs = h1 + (long long)src[e] * HH;
#pragma unroll
    for (int c = 0; c < HH; ++c) atomicAdd(&vacc[c], hs[c] * nm);
  }
}

// ---- seq[t] = relu((vacc + dis0^2 * h1[0]) @ W2 + b2) ---------------------
__global__ void k_seq(const float* __restrict__ vacc, const float* __restrict__ h1,
                      const float* __restrict__ dis, const float* __restrict__ W2,
                      const float* __restrict__ b2, float* __restrict__ seqt) {
  __shared__ float vt[HH];
  int c = threadIdx.x;
  float d0 = dis[0];
  vt[c] = vacc[c] + d0 * d0 * h1[c];
  __syncthreads();
  float s = b2[c];
#pragma unroll 8
  for (int k = 0; k < HH; ++k) s += vt[k] * W2[k * HH + c];
  seqt[c] = s > 0.f ? s : 0.f;
}

// ---- GRU over T steps + 8-head sigmoid, single block of 192 threads -------
__global__ void k_gru_head(const float* __restrict__ seq, const float* __restrict__ Wih,
                           const float* __restrict__ Whh, const float* __restrict__ bih,
                           const float* __restrict__ bhh, const float* __restrict__ headW,
                           const float* __restrict__ headb, float* __restrict__ out) {
  __shared__ float h[HH], xt[HH], gi[3 * HH], gh[3 * HH];
  int j = threadIdx.x;  // 0..191
  if (j < HH) h[j] = 0.f;
  __syncthreads();
  for (int t = 0; t < T_STEPS; ++t) {
    if (j < HH) xt[j] = seq[t * HH + j];
    __syncthreads();
    float a = bih[j], b = bhh[j];
#pragma unroll 8
    for (int k = 0; k < HH; ++k) {
      a += xt[k] * Wih[j * HH + k];
      b += h[k]  * Whh[j * HH + k];
    }
    gi[j] = a;
    gh[j] = b;
    __syncthreads();
    if (j < HH) {
      float r  = 1.f / (1.f + expf(-(gi[j] + gh[j])));
      float z  = 1.f / (1.f + expf(-(gi[HH + j] + gh[HH + j])));
      float nn = tanhf(gi[2 * HH + j] + r * gh[2 * HH + j]);
      h[j] = (1.f - z) * nn + z * h[j];
    }
    __syncthreads();
  }
  if (j < 8) {
    float s = headb[j];
#pragma unroll 8
    for (int k = 0; k < HH; ++k) s += h[k] * headW[j * HH + k];
    out[j] = 1.f / (1.f + expf(-s));
  }
}

// ---------------------------------------------------------------------------
extern "C" void kernel_launch(void* const* d_in, const int* in_sizes, int n_in,
                              void* d_out, int out_size, void* d_ws, size_t ws_size,
                              hipStream_t stream) {
  const float* x     = (const float*)d_in[0];   // [T,N,128]
  const int*   eidx  = (const int*)  d_in[1];   // [T,2,E]
  const float* ew    = (const float*)d_in[2];   // [T,E]
  const float* W1    = (const float*)d_in[3];   // [128,64]
  const float* b1    = (const float*)d_in[4];   // [64]
  const float* W2    = (const float*)d_in[5];   // [64,64]
  const float* b2    = (const float*)d_in[6];   // [64]
  const float* Wih   = (const float*)d_in[7];   // [192,64]
  const float* Whh   = (const float*)d_in[8];   // [192,64]
  const float* bih   = (const float*)d_in[9];   // [192]
  const float* bhh   = (const float*)d_in[10];  // [192]
  const float* headW = (const float*)d_in[11];  // [8,64]
  const float* headb = (const float*)d_in[12];  // [8]
  float* out = (float*)d_out;                   // [8]

  // workspace carve-up (floats)
  float* ws   = (float*)d_ws;
  float* deg  = ws;                          // N   (becomes dis in place)
  float* nrm  = deg + NN;                    // E
  float* h1g  = nrm + EE;                    // N*H  (x@W1)
  float* agg  = h1g + (size_t)NN * HH;       // N*H  (becomes relu'd h1)
  float* vacc = agg + (size_t)NN * HH;       // H
  float* seq  = vacc + HH;                   // T*H

  const int thr = 256;
  const int gN   = (NN + thr - 1) / thr;
  const int gNH  = (NN * HH + thr - 1) / thr;
  const int gE   = (EE + thr - 1) / thr;
  const int gE64 = (int)(((long long)EE * HH + thr - 1) / thr);
  const int gGemm = ((NN / 16) + 3) / 4;     // 4 waves of 32 per 128-thr block

  for (int t = 0; t < T_STEPS; ++t) {
    const float* xt  = x + (size_t)t * NN * FIN;
    const int*   src = eidx + (size_t)t * 2 * EE;
    const int*   dst = src + EE;
    const float* ewt = ew + (size_t)t * EE;

    k_fill<<<gN, thr, 0, stream>>>(deg, 1.0f, NN);          // self-loop weight
    k_fill<<<gNH, thr, 0, stream>>>(agg, 0.0f, NN * HH);
    k_fill<<<1, HH, 0, stream>>>(vacc, 0.0f, HH);

    k_deg<<<gE, thr, 0, stream>>>(dst, ewt, deg, EE);
    k_rsqrt<<<gN, thr, 0, stream>>>(deg, NN);               // deg -> dis
    k_norm<<<gE, thr, 0, stream>>>(src, dst, ewt, deg, nrm, EE);

    k_gemm_wmma<<<gGemm, 128, 0, stream>>>(xt, W1, h1g, NN, FIN);

    k_scatter<<<gE64, thr, 0, stream>>>(h1g, src, dst, nrm, agg, EE);
    k_fin1<<<gNH, thr, 0, stream>>>(agg, h1g, deg, b1, NN); // agg -> h1

    k_red0<<<gE, thr, 0, stream>>>(agg, src, dst, nrm, vacc, EE);
    k_seq<<<1, HH, 0, stream>>>(vacc, agg, deg, W2, b2, seq + t * HH);
  }

  k_gru_head<<<1, 3 * HH, 0, stream>>>(seq, Wih, Whh, bih, bhh, headW, headb, out);
}